// LengthRegulator_24180665876557
// MI455X (gfx1250) — compile-verified
//
#include <hip/hip_runtime.h>
#include <stdint.h>

// Problem constants (fixed by the reference): N=16, L=512, D=256, T=6144.
#define LR_N 16
#define LR_L 512
#define LR_D 256
#define LR_T 6144
#define ROWS_PER_BLOCK 64   // rows of output per workgroup
#define ROWS_PER_WAVE  8    // 8 waves * 8 rows = 64

typedef __attribute__((ext_vector_type(4))) unsigned int uint32x4;
typedef __attribute__((ext_vector_type(8))) int          int32x8;
typedef __attribute__((ext_vector_type(4))) int          int32x4;
typedef __attribute__((ext_vector_type(4))) float        f32x4;   // native vec

// ---------------------------------------------------------------------------
// Kernel 1: per-batch inclusive scan of durations + constant 9.0 fill.
// 16 blocks x 512 threads (16 waves). Hillis-Steele scan in LDS.
// ---------------------------------------------------------------------------
__global__ __launch_bounds__(LR_L) void lr_scan_kernel(
    const int* __restrict__ target,   // (N, L)
    int*       __restrict__ csum_ws,  // (N, L) workspace
    float*     __restrict__ dur_out)  // (N, L) output #2
{
    __shared__ int s[LR_L];
    const int n = blockIdx.x;
    const int i = threadIdx.x;

    s[i] = target[n * LR_L + i];
    __syncthreads();

#pragma unroll
    for (int off = 1; off < LR_L; off <<= 1) {
        int t = (i >= off) ? s[i - off] : 0;
        __syncthreads();
        s[i] += t;
        __syncthreads();
    }

    csum_ws[n * LR_L + i] = s[i];                       // re-read by kernel 2: RT
    __builtin_nontemporal_store(9.0f, &dur_out[n * LR_L + i]);  // pure stream: NT
}

// ---------------------------------------------------------------------------
// Kernel 2: gather. grid = (T/ROWS_PER_BLOCK, N), 256 threads (8 waves).
// Wave 0 DMAs the batch cumsum table (512 x i32 = 2KB) into LDS via the
// Tensor Data Mover; every wave then processes ROWS_PER_WAVE output rows:
// uniform upper_bound binary search in LDS + B128 row copy / zero fill.
// Output stores are non-temporal: 100.7 MB write-only stream must not evict
// the L2-resident x working set (8 MB, re-read ~12x).
// ---------------------------------------------------------------------------
__global__ __launch_bounds__(256) void lr_gather_kernel(
    const float* __restrict__ x,       // (N, L, D)
    const int*   __restrict__ csum,    // (N, L) from kernel 1
    float*       __restrict__ out)     // (N, T, D) output #1
{
    __shared__ int s_csum[LR_L];

    const int n    = blockIdx.y;
    const int tid  = threadIdx.x;
    const int wave = tid >> 5;
    const int lane = tid & 31;

#if defined(__has_builtin) && __has_builtin(__builtin_amdgcn_tensor_load_to_lds)
    if (tid < 32) {  // wave-uniform: only wave 0 issues the TDM op
        // D# group 0: count=1 | lds_addr | global_addr(57b) | type=2
        const unsigned long long ga =
            (unsigned long long)(uintptr_t)(csum + (size_t)n * LR_L);
        // Flat shared pointer: low 32 bits are the LDS byte offset
        // (see ISA 10.2).
        const unsigned lds_off = (unsigned)(uintptr_t)(&s_csum[0]);

        uint32x4 g0 = {0u, 0u, 0u, 0u};
        g0[0] = 0x1u;                                        // count = 1 valid D#
        g0[1] = lds_off;                                     // lds_addr (bytes)
        g0[2] = (unsigned)(ga & 0xFFFFFFFFu);                // global_addr[31:0]
        g0[3] = (unsigned)((ga >> 32) & 0x01FFFFFFu)         // global_addr[56:32]
              | (2u << 30);                                  // type = 2 ("image")

        // D# group 1: data_size=4B; 1-D tensor: dim0=512, tile0=512, stride=512
        int32x8 g1 = {0, 0, 0, 0, 0, 0, 0, 0};
        g1[0] = (int)(2u << 16);         // data_size = 2 -> 4 bytes; wg_mask = 0
        g1[1] = (int)(512u << 16);       // tensor_dim0[15:0] = 512
        g1[2] = (int)(1u << 16);         // tensor_dim1 = 1
        g1[3] = (int)(512u << 16);       // tile_dim0 = 512
        g1[5] = 512;                     // tensor_dim0_stride = 512 elements

        int32x4 gz4 = {0, 0, 0, 0};            // groups 2/3 unused (<=2D tensor)
        int32x8 gz8 = {0, 0, 0, 0, 0, 0, 0, 0};
        __builtin_amdgcn_tensor_load_to_lds(g0, g1, gz4, gz4, gz8, 0);
        __builtin_amdgcn_s_wait_tensorcnt(0);
    }
#else
    for (int i = tid; i < LR_L; i += 256)
        s_csum[i] = csum[n * LR_L + i];
#endif
    __syncthreads();

    const int total = s_csum[LR_L - 1];

    const f32x4* __restrict__ xrow =
        (const f32x4*)(x + (size_t)n * LR_L * LR_D);           // D/4 = 64 f4/row
    f32x4* __restrict__ orow =
        (f32x4*)(out + (size_t)n * LR_T * LR_D);

    const int row_base = blockIdx.x * ROWS_PER_BLOCK + wave * ROWS_PER_WAVE;

#pragma unroll
    for (int r = 0; r < ROWS_PER_WAVE; ++r) {
        const int t = row_base + r;

        // upper_bound: first i with csum[i] > t  (searchsorted side='right')
        int lo = 0, hi = LR_L;
#pragma unroll
        for (int step = 0; step < 9; ++step) {   // log2(512) = 9 iterations
            const int mid = (lo + hi) >> 1;
            if (t < s_csum[mid]) hi = mid; else lo = mid + 1;
        }
        int idx = (lo < LR_L) ? lo : (LR_L - 1);

        f32x4 a, b;
        if (t < total) {                          // wave-uniform branch
            const f32x4* src = xrow + (size_t)idx * (LR_D / 4);
            // idx is monotone non-decreasing in t: pull the next row toward L0
            __builtin_prefetch(xrow + (size_t)((idx < LR_L - 1) ? idx + 1 : idx)
                                          * (LR_D / 4), 0, 1);
            a = src[lane];
            b = src[lane + 32];
        } else {
            a = (f32x4){0.f, 0.f, 0.f, 0.f};
            b = a;
        }
        f32x4* dst = orow + (size_t)t * (LR_D / 4);
        __builtin_nontemporal_store(a, &dst[lane]);        // streaming NT store
        __builtin_nontemporal_store(b, &dst[lane + 32]);
    }
}

// ---------------------------------------------------------------------------
// Host-side launcher.
// Inputs (setup_inputs order): emb_out f32 (unused), x f32 (N,L,D),
// target i32 (N,L), max_length i32 scalar (fixed 6144).
// d_out: output (N,T,D) f32 then duration_predictor_output (N,L) f32.
// ---------------------------------------------------------------------------
extern "C" void kernel_launch(void* const* d_in, const int* in_sizes, int n_in,
                              void* d_out, int out_size, void* d_ws, size_t ws_size,
                              hipStream_t stream)
{
    (void)in_sizes; (void)n_in; (void)out_size; (void)ws_size;

    const float* x      = (const float*)d_in[1];
    const int*   target = (const int*)d_in[2];

    float* out = (float*)d_out;                                   // (N,T,D)
    float* dur = out + (size_t)LR_N * LR_T * LR_D;                // (N,L)
    int*   csum_ws = (int*)d_ws;                                  // (N,L) ints

    lr_scan_kernel<<<dim3(LR_N), dim3(LR_L), 0, stream>>>(target, csum_ws, dur);

    dim3 grid(LR_T / ROWS_PER_BLOCK, LR_N);   // (96, 16)
    lr_gather_kernel<<<grid, dim3(256), 0, stream>>>(x, csum_ws, out);
}